// NConvToBatch2D_30983894073618
// MI455X (gfx1250) — compile-verified
//
#include <hip/hip_runtime.h>

typedef __attribute__((ext_vector_type(2))) float v2f;
typedef __attribute__((ext_vector_type(8))) float v8f;

#define NCONV 4
#define CIN   64
#define FILT  128
#define MPIX  (8 * 256 * 256)       // 524288 pixels
#define BLOCK_PIX 32                // pixels staged per block
#define LDS_STRIDE 68               // 64 + 4 pad floats: kills 16-way bank conflicts

// Each block: 256 threads = 8 wave32. Wave w handles branch n = w>>1 over a
// 16-pixel subtile (w&1). Each wave computes a 16x128 fp32 tile with
// v_wmma_f32_16x16x4_f32 (16 K-steps x 8 N-tiles = 128 WMMAs).
__global__ __launch_bounds__(256) void nconv_wmma_kernel(
    const float* __restrict__ x,      // [MPIX][64]
    const float* __restrict__ Wt,     // [4][64][128]
    const float* __restrict__ bias,   // [4][128]
    float* __restrict__ out)          // [4][MPIX][128]
{
    __shared__ float xs[BLOCK_PIX * LDS_STRIDE];

    const int tid      = threadIdx.x;
    const int blockPix = blockIdx.x * BLOCK_PIX;

    // --- Stage x tile: 32 pixels x 64 ch = 2048 floats, 2 x float4 per thread.
    {
        const float4* src = (const float4*)(x + (size_t)blockPix * CIN);
        #pragma unroll
        for (int i = 0; i < 2; ++i) {
            int g = tid + i * 256;            // float4 index within tile (512 total)
            float4 v = src[g];
            int p = (g * 4) / CIN;            // pixel in tile
            int c = (g * 4) % CIN;            // channel
            *(float4*)(&xs[p * LDS_STRIDE + c]) = v;
        }
    }
    __syncthreads();

    const int wave   = tid >> 5;
    const int lane   = tid & 31;
    const int n      = wave >> 1;     // conv branch 0..3
    const int msub   = wave & 1;      // 16-pixel subtile 0..1
    const int laneHi = lane >> 4;     // 0: K base +0/+1, 1: K base +2/+3
    const int laneLo = lane & 15;

    const float* Wn = Wt + (size_t)n * CIN * FILT;

    v8f acc[8];
    #pragma unroll
    for (int nt = 0; nt < 8; ++nt) acc[nt] = (v8f)0.0f;

    // A-fragment row in LDS: lanes 0-15 carry M=row (ISA 16x4 f32 layout).
    const int arow = msub * 16 + laneLo;
    const float* aBase = &xs[arow * LDS_STRIDE];

    // --- K loop: 16 steps of K=4.
    #pragma unroll 4
    for (int t = 0; t < 16; ++t) {
        const int k = 4 * t + 2 * laneHi;
        v2f a;
        a.x = aBase[k];
        a.y = aBase[k + 1];

        const float* wk = Wn + (size_t)k * FILT;   // row k of this branch's weights
        #pragma unroll
        for (int nt = 0; nt < 8; ++nt) {
            v2f b;
            b.x = wk[nt * 16 + laneLo];            // K = k
            b.y = wk[FILT + nt * 16 + laneLo];     // K = k+1
            acc[nt] = __builtin_amdgcn_wmma_f32_16x16x4_f32(
                /*neg_a=*/false, a, /*neg_b=*/false, b,
                /*c_mod=*/(short)0, acc[nt],
                /*reuse_a=*/false, /*reuse_b=*/false);
        }
    }

    // --- Epilogue: bias + ReLU + store.
    // C/D layout: VGPR r -> lanes 0-15: (M=r, N=laneLo), lanes 16-31: (M=r+8, N=laneLo).
    const int rowBase = blockPix + msub * 16;
    float* outn = out + (size_t)n * MPIX * FILT;
    #pragma unroll
    for (int nt = 0; nt < 8; ++nt) {
        const float bv = bias[n * FILT + nt * 16 + laneLo];
        #pragma unroll
        for (int r = 0; r < 8; ++r) {
            const int m = rowBase + r + laneHi * 8;
            float v = acc[nt][r] + bv;
            v = v > 0.0f ? v : 0.0f;
            outn[(size_t)m * FILT + nt * 16 + laneLo] = v;
        }
    }
}

extern "C" void kernel_launch(void* const* d_in, const int* in_sizes, int n_in,
                              void* d_out, int out_size, void* d_ws, size_t ws_size,
                              hipStream_t stream) {
    const float* x    = (const float*)d_in[0];   // 8*256*256*64 fp32
    const float* Wt   = (const float*)d_in[1];   // 4*64*128 fp32
    const float* bias = (const float*)d_in[2];   // 4*128 fp32
    float* out        = (float*)d_out;           // 4*8*256*256*128 fp32

    const int grid = MPIX / BLOCK_PIX;           // 16384 blocks
    nconv_wmma_kernel<<<grid, 256, 0, stream>>>(x, Wt, bias, out);
}